// GIN_76390288327116
// MI455X (gfx1250) — compile-verified
//
#include <hip/hip_runtime.h>

typedef __attribute__((ext_vector_type(2))) float v2f;
typedef __attribute__((ext_vector_type(8))) float v8f;

#define NN 100000
#define NCH 4

__device__ __forceinline__ v8f wmma_f32(v2f a, v2f b, v8f c) {
  // V_WMMA_F32_16X16X4_F32 : D = A(16x4) * B(4x16) + C(16x16), fp32 throughout
  return __builtin_amdgcn_wmma_f32_16x16x4_f32(false, a, false, b, (short)0, c,
                                               false, false);
}

// ---------------------------------------------------------------------------
// Weight repack: W[K][C] row-major -> Wp[k/4][C][4]  (k-quads contiguous per
// column). Lets each lane fetch its (kk, kk+1) B-fragment pair as ONE b64
// load, and a wave's 32 lanes cover one contiguous 256B span per k-step.
// blockIdx.y = matrix index (batched over identical-shape matrices).
// ---------------------------------------------------------------------------
__global__ __launch_bounds__(256) void repack_w(
    const float* __restrict__ W, float* __restrict__ Wp, int K, int C)
{
  const size_t mat = (size_t)blockIdx.y * K * C;
  size_t idx = (size_t)blockIdx.x * 256 + threadIdx.x;
  if (idx >= (size_t)K * C) return;
  int k   = (int)(idx / C);
  int col = (int)(idx % C);
  Wp[mat + (((size_t)(k >> 2) * C + col) << 2) + (k & 3)] = W[mat + idx];
}

// ---------------------------------------------------------------------------
// Strided row copy of 128-float rows (float4 per thread-iteration).
// Used for: xcat[:,0:128] = x   and   agg = h  (GIN self term).
// ---------------------------------------------------------------------------
__global__ __launch_bounds__(256) void copy_rows128(
    const float* __restrict__ in, int ldi,
    float* __restrict__ outp, int ldo, int nRows)
{
  size_t idx = (size_t)blockIdx.x * 256 + threadIdx.x;   // one float4 each
  if (idx >= (size_t)nRows * 32) return;
  size_t row = idx >> 5;
  int c4 = (int)(idx & 31) << 2;
  const float4 v = *(const float4*)(in + row * (size_t)ldi + c4);
  *(float4*)(outp + row * (size_t)ldo + c4) = v;
}

// ---------------------------------------------------------------------------
// Edge scatter-add: agg[dst] += h[src].  32 threads per edge, float4 gather,
// fp32 HW atomics (global_atomic_add_f32 via unsafeAtomicAdd).
// ---------------------------------------------------------------------------
__global__ __launch_bounds__(256) void gin_scatter(
    const float* __restrict__ feat, int ldf,
    const int* __restrict__ srcIdx, const int* __restrict__ dstIdx,
    float* __restrict__ agg, long long E)
{
  long long t = (long long)blockIdx.x * 256 + threadIdx.x;
  long long e = t >> 5;
  if (e >= E) return;
  int g = (int)(t & 31);
  int s = srcIdx[e];
  int d = dstIdx[e];
  const float4 v = *(const float4*)(feat + (size_t)s * ldf + g * 4);
  float* p = agg + (size_t)d * 128 + g * 4;
  unsafeAtomicAdd(p + 0, v.x);
  unsafeAtomicAdd(p + 1, v.y);
  unsafeAtomicAdd(p + 2, v.z);
  unsafeAtomicAdd(p + 3, v.w);
}

// ---------------------------------------------------------------------------
// Fused 3x (Linear 128x128 + bias + ReLU) on a 32-node tile, fp32 WMMA.
// 8 waves: wave w owns column tile [16w,16w+16) for both 16-row strips.
// LDS row stride 132 (132 % 64 == 4 -> conflict-free fragment reads).
// Weights come pre-packed (k-quads contiguous) -> single b64 per B fragment.
// Output written into xcat at column offset colOff (ld 512) == concat.
// ---------------------------------------------------------------------------
__global__ __launch_bounds__(256) void gin_mlp3(
    const float* __restrict__ agg,   // [N,128]
    const float* __restrict__ Wp,    // [3][128/4][128][4] packed (this layer)
    const float* __restrict__ Bv,    // [3,128]
    float* __restrict__ xcat, int colOff)
{
  __shared__ float bufA[32][132];
  __shared__ float bufB[32][132];
  const int tid  = threadIdx.x;
  const int lane = tid & 31;
  const int w    = tid >> 5;
  const size_t base = (size_t)blockIdx.x * 32;

  for (int i = tid; i < 32 * 32; i += 256) {       // 32x128 in float4 units
    int row = i >> 5, c4 = (i & 31) << 2;
    const float4 v = *(const float4*)(agg + (base + row) * 128 + c4);
    bufA[row][c4] = v.x; bufA[row][c4 + 1] = v.y;
    bufA[row][c4 + 2] = v.z; bufA[row][c4 + 3] = v.w;
  }
  __syncthreads();

  const int half = lane >> 4;       // lane half selects K {0,1}/{2,3} pattern
  const int n16  = lane & 15;
  const int cb   = w * 16;          // column tile base

  for (int layer = 0; layer < 3; ++layer) {
    const float* Wk = Wp + (size_t)layer * 128 * 128;
    float (*in)[132]   = (layer & 1) ? bufB : bufA;
    float (*outb)[132] = (layer & 1) ? bufA : bufB;
    v8f acc0 = {}, acc1 = {};
    for (int k = 0; k < 128; k += 4) {
      const int kk = k + half * 2;
      const v2f b = *(const v2f*)(Wk + (((size_t)(k >> 2) * 128 + cb + n16) << 2)
                                     + half * 2);
      v2f a0; a0.x = in[n16][kk];      a0.y = in[n16][kk + 1];
      v2f a1; a1.x = in[n16 + 16][kk]; a1.y = in[n16 + 16][kk + 1];
      acc0 = wmma_f32(a0, b, acc0);
      acc1 = wmma_f32(a1, b, acc1);
    }
    const float bias = Bv[layer * 128 + cb + n16];
    #pragma unroll
    for (int j = 0; j < 8; ++j) {
      float r0 = acc0[j] + bias; r0 = r0 > 0.f ? r0 : 0.f;
      float r1 = acc1[j] + bias; r1 = r1 > 0.f ? r1 : 0.f;
      outb[j + half * 8][cb + n16]      = r0;
      outb[16 + j + half * 8][cb + n16] = r1;
    }
    __syncthreads();
  }
  // layer 2 wrote bufB
  for (int i = tid; i < 32 * 32; i += 256) {
    int row = i >> 5, c4 = (i & 31) << 2;
    float4 v = make_float4(bufB[row][c4], bufB[row][c4 + 1],
                           bufB[row][c4 + 2], bufB[row][c4 + 3]);
    *(float4*)(xcat + (base + row) * 512 + colOff + c4) = v;
  }
}

// ---------------------------------------------------------------------------
// Channel MLP: 512 -> 256 -> 256 -> 1 (ReLU everywhere), one block per
// (16-node tile, channel). Wave w owns column tiles {w, w+8}.
// Weights pre-packed as above.
// ---------------------------------------------------------------------------
__global__ __launch_bounds__(256) void gin_chan(
    const float* __restrict__ xcat,
    const float* __restrict__ W1p, const float* __restrict__ B1,
    const float* __restrict__ W2p, const float* __restrict__ B2,
    const float* __restrict__ W3,  const float* __restrict__ B3,
    float* __restrict__ outp)
{
  __shared__ float bufA[16][516];   // xcat tile, then reused for h2 (cols 0..255)
  __shared__ float bufB[16][260];   // h1
  const int tid  = threadIdx.x;
  const int lane = tid & 31;
  const int w    = tid >> 5;
  const int c    = blockIdx.y;
  const size_t base = (size_t)blockIdx.x * 16;

  for (int i = tid; i < 16 * 128; i += 256) {      // 16x512 in float4 units
    int row = i >> 7, c4 = (i & 127) << 2;
    const float4 v = *(const float4*)(xcat + (base + row) * 512 + c4);
    bufA[row][c4] = v.x; bufA[row][c4 + 1] = v.y;
    bufA[row][c4 + 2] = v.z; bufA[row][c4 + 3] = v.w;
  }
  __syncthreads();

  const int half = lane >> 4;
  const int n16  = lane & 15;
  const int ct0  = w * 16;
  const int ct1  = (w + 8) * 16;

  // ---- layer 1: [16,512] @ [512,256] ----
  {
    const float* W1c = W1p + (size_t)c * 512 * 256;
    v8f acc0 = {}, acc1 = {};
    for (int k = 0; k < 512; k += 4) {
      const int kk = k + half * 2;
      v2f a; a.x = bufA[n16][kk]; a.y = bufA[n16][kk + 1];
      const size_t kb = ((size_t)(k >> 2) * 256) << 2;
      const v2f b0 = *(const v2f*)(W1c + kb + ((ct0 + n16) << 2) + half * 2);
      acc0 = wmma_f32(a, b0, acc0);
      const v2f b1 = *(const v2f*)(W1c + kb + ((ct1 + n16) << 2) + half * 2);
      acc1 = wmma_f32(a, b1, acc1);
    }
    const float bias0 = B1[c * 256 + ct0 + n16];
    const float bias1 = B1[c * 256 + ct1 + n16];
    #pragma unroll
    for (int j = 0; j < 8; ++j) {
      float r0 = acc0[j] + bias0; r0 = r0 > 0.f ? r0 : 0.f;
      float r1 = acc1[j] + bias1; r1 = r1 > 0.f ? r1 : 0.f;
      bufB[j + half * 8][ct0 + n16] = r0;
      bufB[j + half * 8][ct1 + n16] = r1;
    }
  }
  __syncthreads();

  // ---- layer 2: [16,256] @ [256,256] ----
  {
    const float* W2c = W2p + (size_t)c * 256 * 256;
    v8f acc0 = {}, acc1 = {};
    for (int k = 0; k < 256; k += 4) {
      const int kk = k + half * 2;
      v2f a; a.x = bufB[n16][kk]; a.y = bufB[n16][kk + 1];
      const size_t kb = ((size_t)(k >> 2) * 256) << 2;
      const v2f b0 = *(const v2f*)(W2c + kb + ((ct0 + n16) << 2) + half * 2);
      acc0 = wmma_f32(a, b0, acc0);
      const v2f b1 = *(const v2f*)(W2c + kb + ((ct1 + n16) << 2) + half * 2);
      acc1 = wmma_f32(a, b1, acc1);
    }
    const float bias0 = B2[c * 256 + ct0 + n16];
    const float bias1 = B2[c * 256 + ct1 + n16];
    #pragma unroll
    for (int j = 0; j < 8; ++j) {
      float r0 = acc0[j] + bias0; r0 = r0 > 0.f ? r0 : 0.f;
      float r1 = acc1[j] + bias1; r1 = r1 > 0.f ? r1 : 0.f;
      bufA[j + half * 8][ct0 + n16] = r0;
      bufA[j + half * 8][ct1 + n16] = r1;
    }
  }
  __syncthreads();

  // ---- layer 3: [16,256] @ [256,1] (tiny -> dot product on wave 0) ----
  if (w == 0 && lane < 16) {
    const float* w3 = W3 + (size_t)c * 256;
    float sum = 0.f;
    for (int k = 0; k < 256; ++k) sum += bufA[lane][k] * w3[k];
    sum += B3[c];
    sum = sum > 0.f ? sum : 0.f;
    outp[(base + lane) * NCH + c] = sum;
  }
}

// ---------------------------------------------------------------------------
extern "C" void kernel_launch(void* const* d_in, const int* in_sizes, int n_in,
                              void* d_out, int out_size, void* d_ws, size_t ws_size,
                              hipStream_t stream)
{
  (void)n_in; (void)out_size; (void)ws_size;
  const float* x      = (const float*)d_in[0];
  const int*   ei     = (const int*)d_in[1];
  const float* convW  = (const float*)d_in[2];
  const float* convB  = (const float*)d_in[3];
  const float* chanW1 = (const float*)d_in[4];
  const float* chanB1 = (const float*)d_in[5];
  const float* chanW2 = (const float*)d_in[6];
  const float* chanB2 = (const float*)d_in[7];
  const float* chanW3 = (const float*)d_in[8];
  const float* chanB3 = (const float*)d_in[9];
  float* outp = (float*)d_out;

  const int N = NN;
  const long long E = (long long)in_sizes[1] / 2;

  // workspace layout
  float* xcat   = (float*)d_ws;                       // [N,512]
  float* agg    = xcat + (size_t)N * 512;             // [N,128]
  float* convWp = agg + (size_t)N * 128;              // 9 x 128x128 packed
  float* W1p    = convWp + (size_t)9 * 128 * 128;     // 4 x 512x256 packed
  float* W2p    = W1p + (size_t)4 * 512 * 256;        // 4 x 256x256 packed

  // pre-pack all GEMM weights into k-quad layout (tiny, deterministic)
  repack_w<<<dim3((128 * 128) / 256, 9), 256, 0, stream>>>(convW, convWp, 128, 128);
  repack_w<<<dim3((512 * 256) / 256, 4), 256, 0, stream>>>(chanW1, W1p, 512, 256);
  repack_w<<<dim3((256 * 256) / 256, 4), 256, 0, stream>>>(chanW2, W2p, 256, 256);

  const int cpBlocks = (int)(((size_t)N * 32 + 255) / 256);
  const int scBlocks = (int)((E * 32 + 255) / 256);

  // feats[0] = x  (concat slot 0)
  copy_rows128<<<cpBlocks, 256, 0, stream>>>(x, 128, xcat, 512, N);

  for (int l = 0; l < 3; ++l) {
    const float* h = (l == 0) ? x : (xcat + (size_t)l * 128);
    const int ldh  = (l == 0) ? 128 : 512;
    // agg = h  (self term, eps = 0)
    copy_rows128<<<cpBlocks, 256, 0, stream>>>(h, ldh, agg, 128, N);
    // agg[dst] += h[src]
    gin_scatter<<<scBlocks, 256, 0, stream>>>(h, ldh, ei, ei + E, agg, E);
    // 3x (Linear+ReLU) -> concat slot l+1
    gin_mlp3<<<N / 32, 256, 0, stream>>>(agg,
                                         convWp + (size_t)l * 3 * 128 * 128,
                                         convB + l * 3 * 128,
                                         xcat, (l + 1) * 128);
  }

  gin_chan<<<dim3(N / 16, NCH), 256, 0, stream>>>(xcat, W1p, chanB1,
                                                  W2p, chanB2,
                                                  chanW3, chanB3, outp);
}